// OriEmbeddingBlock_62148176773667
// MI455X (gfx1250) — compile-verified
//
#include <hip/hip_runtime.h>

// out[N,1152] = T[N,12] @ W[12,1152]   (K padded 9 -> 12)
//   T rows: [ r^2/sqrt3, 0,0,0, 2a*xy, 2a*yz, (a/sqrt3)(2z^2-x^2-y^2), 2a*zx, a(x^2-y^2), 0,0,0 ]
//   e3nn basis (component 0=y, 1=z, 2=x), a = 1/sqrt(10). The 1e path (K=1..3) is
//   identically zero (antisymmetric CG on a symmetric outer product).
//   W columns each have exactly ONE nonzero: cols [0,128): row0 = w[c];
//   cols [128,512): row 1+cc%3 = sqrt3*w[128+cc/3]; cols [512,1152): row 4+cc%5 = sqrt5*w[256+cc/5].
//
// MI455X reasoning: 461 MB f32 output vs ~1 MB input -> pure store-BW bound
// (~20 us floor @ 23.3 TB/s). V_WMMA_F32_16X16X4_F32, 3 chained (K=12), per 16x16
// output tile. Column tile is grid-parallel so the sparse B fragment is built ONCE
// per wave, branchlessly (cndmask only -> EXEC stays all-ones for WMMA); the wave
// then streams row tiles: 1 coalesced ori load + ~15 VALU + 3 WMMA + 8 NT stores.
// Non-temporal stores: output exceeds the 192 MB L2 and is never re-read.

typedef __attribute__((ext_vector_type(2))) float v2f;
typedef __attribute__((ext_vector_type(8))) float v8f;

#define COLS 1152
#define CTILES 72          // 1152 / 16 column tiles
#define CHUNK 32           // row tiles (of 16 rows) per wave

__global__ __launch_bounds__(256)
void tp_wmma_kernel(const float* __restrict__ ori,
                    const float* __restrict__ weights,
                    float* __restrict__ out,
                    int full_tiles, int nchunks) {
  const int lane = threadIdx.x & 31;
  const int wid  = blockIdx.x * 8 + (threadIdx.x >> 5);
  if (wid >= CTILES * nchunks) return;            // wave-uniform exit

  const int j     = wid % CTILES;                 // column tile
  const int chunk = wid / CTILES;                 // row-tile chunk
  const int t0    = chunk * CHUNK;
  const int t1    = (t0 + CHUNK < full_tiles) ? t0 + CHUNK : full_tiles;

  const int m    = lane & 15;
  const int half = lane >> 4;

  // ---- B fragment for this wave's 16 columns, built once, branchless ----
  // B layout (mirror of documented A layout): VGPR.x holds K = kbase + (half?2:0),
  // VGPR.y holds K = kbase + (half?3:1); N = 16*j + (lane&15) for both halves.
  const float s3 = 1.7320508075688772f, s5 = 2.2360679774997896f;
  const int col = j * 16 + m;
  const bool in1 = (col >= 128) & (col < 512);
  const bool in2 = (col >= 512);
  const int cc1 = col - 128, cc2 = col - 512;
  const int r     = in2 ? 4 + cc2 % 5 : (in1 ? 1 + cc1 % 3 : 0);
  const int widx  = in2 ? 256 + cc2 / 5 : (in1 ? 128 + cc1 / 3 : col);
  const float scl = in2 ? s5 : (in1 ? s3 : 1.0f);
  const float val = scl * weights[widx];

  const int kx = half ? 2 : 0;
  const int ky = half ? 3 : 1;
  v2f B0, B1, B2;
  B0.x = (r == 0 + kx) ? val : 0.f;  B0.y = (r == 0 + ky) ? val : 0.f;
  B1.x = (r == 4 + kx) ? val : 0.f;  B1.y = (r == 4 + ky) ? val : 0.f;
  B2.x = (r == 8 + kx) ? val : 0.f;  B2.y = (r == 8 + ky) ? val : 0.f;

  const float is3 = 0.57735026918962576f;   // 1/sqrt(3)
  const float a10 = 0.31622776601683794f;   // 1/sqrt(10)

  // ---- stream row tiles ----
  for (int t = t0; t < t1; ++t) {
    const int row0 = t * 16;
    const float* op = ori + (size_t)(row0 + m) * 3;   // 12 B/lane, coalesced per half
    const float o0 = op[0], o1 = op[1], o2 = op[2];   // o0=y, o1=z, o2=x

    const float T0 = (o0*o0 + o1*o1 + o2*o2) * is3;
    const float T4 = 2.f*a10 * o2*o0;                         // 2a*x*y
    const float T5 = 2.f*a10 * o0*o1;                         // 2a*y*z
    const float T6 = (a10*is3) * (2.f*o1*o1 - o2*o2 - o0*o0);
    const float T7 = 2.f*a10 * o1*o2;                         // 2a*z*x
    const float T8 = a10 * (o2*o2 - o0*o0);                   // a(x^2-y^2)

    // A layout (ISA 16x4 f32): lanes 0-15 hold K={0,1}, lanes 16-31 hold K={2,3}.
    v2f A0, A1, A2;
    A0.x = half ? 0.f : T0;  A0.y = 0.f;              // K 0..3 (1e path zero)
    A1.x = half ? T6  : T4;  A1.y = half ? T7 : T5;   // K 4..7
    A2.x = half ? 0.f : T8;  A2.y = 0.f;              // K 8..11 (pad)

    v8f acc = {};
    acc = __builtin_amdgcn_wmma_f32_16x16x4_f32(false, A0, false, B0, (short)0, acc, false, false);
    acc = __builtin_amdgcn_wmma_f32_16x16x4_f32(false, A1, false, B1, (short)0, acc, false, false);
    acc = __builtin_amdgcn_wmma_f32_16x16x4_f32(false, A2, false, B2, (short)0, acc, false, false);

    // D layout: VGPR v -> row row0 + v + half*8, col = 16j + (lane&15).
    float* p = out + (size_t)(row0 + half * 8) * COLS + col;
#pragma unroll
    for (int v = 0; v < 8; ++v)
      __builtin_nontemporal_store(acc[v], p + (size_t)v * COLS);
  }
}

// Tail rows (n % 16) — plain VALU, launched only when needed (never for N=100000).
__global__ void tp_tail_kernel(const float* __restrict__ ori,
                               const float* __restrict__ weights,
                               float* __restrict__ out, int n, int row_start) {
  const float is3 = 0.57735026918962576f, a10 = 0.31622776601683794f;
  const float s5 = 2.2360679774997896f;
  int idx = blockIdx.x * blockDim.x + threadIdx.x;
  const int total = (n - row_start) * COLS;
  for (; idx < total; idx += gridDim.x * blockDim.x) {
    const int row = row_start + idx / COLS;
    const int c   = idx % COLS;
    const float* op = ori + (size_t)row * 3;
    const float o0 = op[0], o1 = op[1], o2 = op[2];
    float res = 0.f;
    if (c < 128) {
      res = (o0*o0 + o1*o1 + o2*o2) * is3 * weights[c];
    } else if (c >= 512) {
      const int cc = c - 512, k = cc % 5, w = cc / 5;
      float t = 0.f;
      if      (k == 0) t = 2.f*a10 * o2*o0;
      else if (k == 1) t = 2.f*a10 * o0*o1;
      else if (k == 2) t = (a10*is3) * (2.f*o1*o1 - o2*o2 - o0*o0);
      else if (k == 3) t = 2.f*a10 * o1*o2;
      else             t = a10 * (o2*o2 - o0*o0);
      res = s5 * t * weights[256 + w];
    }                                   // 1e section stays exactly zero
    out[(size_t)row * COLS + c] = res;
  }
}

extern "C" void kernel_launch(void* const* d_in, const int* in_sizes, int n_in,
                              void* d_out, int out_size, void* d_ws, size_t ws_size,
                              hipStream_t stream) {
  const float* ori     = (const float*)d_in[0];   // [N,3] f32
  const float* weights = (const float*)d_in[1];   // [384] f32
  float* out = (float*)d_out;                     // [N,1152] f32
  const int n = in_sizes[0] / 3;

  const int full_tiles = n / 16;                        // 6250 for N=100000
  if (full_tiles > 0) {
    const int nchunks = (full_tiles + CHUNK - 1) / CHUNK;
    const int nwaves  = CTILES * nchunks;               // 72 * 196 = 14112
    const int nblocks = (nwaves + 7) / 8;
    tp_wmma_kernel<<<nblocks, 256, 0, stream>>>(ori, weights, out, full_tiles, nchunks);
  }
  const int rem_start = full_tiles * 16;
  if (rem_start < n) {
    tp_tail_kernel<<<64, 256, 0, stream>>>(ori, weights, out, n, rem_start);
  }
}